// Receiver_87900800680733
// MI455X (gfx1250) — compile-verified
//
#include <hip/hip_runtime.h>
#include <hip/hip_bf16.h>

#define N_NODES 50000
#define N_EDGES 800000
#define EMB 30
#define HID 80
#define VOCAB_SZ 10
#define GAT_SLOPE 0.2f
#define CONV_SLOPE 0.01f

typedef float v2f __attribute__((ext_vector_type(2)));
typedef float v8f __attribute__((ext_vector_type(8)));

// ---------- helpers ----------
__device__ __forceinline__ unsigned f2mono(float f) {
    unsigned u = __float_as_uint(f);
    return (u & 0x80000000u) ? ~u : (u | 0x80000000u);
}
__device__ __forceinline__ float mono2f(unsigned u) {
    return (u & 0x80000000u) ? __uint_as_float(u & 0x7FFFFFFFu) : __uint_as_float(~u);
}

__global__ void k_fill_f32(float* p, float v, int n) {
    int i = blockIdx.x * blockDim.x + threadIdx.x;
    if (i < n) p[i] = v;
}
__global__ void k_fill_u32(unsigned* p, unsigned v, int n) {
    int i = blockIdx.x * blockDim.x + threadIdx.x;
    if (i < n) p[i] = v;
}

// ---------- layer 1 GEMM via WMMA fp32: z1[N,80] = feat[N,3](pad K=4) @ W1[3,0:80] ----------
__global__ __launch_bounds__(32) void k_z1_wmma(const float* __restrict__ nf, // [N,3]
                                                const float* __restrict__ W1, // gat1_W [3,160]
                                                float* __restrict__ z1) {     // [N,80]
    int tile = blockIdx.x;   // 3125 tiles of 16 rows
    int lane = threadIdx.x;  // wave32, EXEC all ones
    int r = lane & 15;
    int h = lane >> 4;
    int row = tile * 16 + r;
    float hmask = (h == 0) ? 1.f : 0.f;  // K=3 padded: k=3 contributes 0
    // A fragment: lane holds A[M, 2h], A[M, 2h+1]; clamp pad address, mask value
    v2f a;
    a.x = nf[row * 3 + 2 * h];                    // k = 0 or 2 (always valid)
    a.y = nf[row * 3 + (h ? 2 : 1)] * hmask;      // k = 1, or pad k=3 -> 0
    v8f c[5];
#pragma unroll
    for (int t5 = 0; t5 < 5; ++t5) c[t5] = (v8f){};
#pragma unroll
    for (int t5 = 0; t5 < 5; ++t5) {
        int col = t5 * 16 + r;  // 0..79, head 0 of W1 [3,160]
        v2f b;
        b.x = W1[(2 * h) * 160 + col];
        b.y = W1[(h ? 2 : 1) * 160 + col] * hmask;
        c[t5] = __builtin_amdgcn_wmma_f32_16x16x4_f32(false, a, false, b, (short)0, c[t5],
                                                      false, false);
    }
#pragma unroll
    for (int t5 = 0; t5 < 5; ++t5) {
#pragma unroll
        for (int v = 0; v < 8; ++v) {
            int m = v + 8 * h;
            z1[(tile * 16 + m) * HID + t5 * 16 + r] = c[t5][v];
        }
    }
}

// ---------- generic attention dots: el[n] = z[n,:]·al, er[n] = z[n,:]·ar ----------
__global__ void k_elr(const float* __restrict__ z, const float* __restrict__ al,
                      const float* __restrict__ ar, float* __restrict__ el,
                      float* __restrict__ er, int stride, int width) {
    int n = blockIdx.x * blockDim.x + threadIdx.x;
    if (n >= N_NODES) return;
    float accl = 0.f, accr = 0.f;
    for (int j = 0; j < width; ++j) {
        float zv = z[n * stride + j];
        accl += zv * al[j];
        accr += zv * ar[j];
    }
    el[n] = accl;
    er[n] = accr;
}

// ---------- edge kernels (shared by both layers) ----------
__global__ void k_edge_e(const int* __restrict__ src, const int* __restrict__ dst,
                         const float* __restrict__ el, const float* __restrict__ er,
                         float* __restrict__ ebuf, unsigned* __restrict__ emaxu, int E) {
    int e = blockIdx.x * blockDim.x + threadIdx.x;
    if (e >= E) return;
    float v = el[src[e]] + er[dst[e]];
    v = (v >= 0.f) ? v : GAT_SLOPE * v;
    ebuf[e] = v;
    atomicMax(&emaxu[dst[e]], f2mono(v));
}

__global__ void k_edge_exp(const int* __restrict__ dst, float* __restrict__ ebuf,
                           const unsigned* __restrict__ emaxu, float* __restrict__ denom, int E) {
    int e = blockIdx.x * blockDim.x + threadIdx.x;
    if (e >= E) return;
    int d = dst[e];
    float ex = expf(ebuf[e] - mono2f(emaxu[d]));
    ebuf[e] = ex;
    atomicAdd(&denom[d], ex);
}

__global__ void k_alpha(const int* __restrict__ dst, float* __restrict__ ebuf,
                        const float* __restrict__ denom, int E) {
    int e = blockIdx.x * blockDim.x + threadIdx.x;
    if (e >= E) return;
    ebuf[e] = ebuf[e] / denom[dst[e]];
}

// ---------- layer 1 scatter: rst[dst,:] += z1[src,:]*alpha ----------
__global__ void k_scatter(const int* __restrict__ src, const int* __restrict__ dst,
                          const float* __restrict__ alpha, const float* __restrict__ z1,
                          float* __restrict__ rst, int total) {
    int idx = blockIdx.x * blockDim.x + threadIdx.x;
    if (idx >= total) return;
    int e = idx / HID;
    int f = idx - e * HID;
    atomicAdd(&rst[dst[e] * HID + f], z1[src[e] * HID + f] * alpha[e]);
}

__global__ void k_bias_relu(float* __restrict__ g, const float* __restrict__ b, int total) {
    int idx = blockIdx.x * blockDim.x + threadIdx.x;
    if (idx >= total) return;
    float v = g[idx] + b[idx % HID];
    g[idx] = v > 0.f ? v : 0.f;
}

// ---------- layer 2 GEMM via WMMA fp32: z2[N,30] = g1[N,80] @ W[80,0:30], padded to 32 cols ----------
__global__ __launch_bounds__(32) void k_z2_wmma(const float* __restrict__ g1,
                                                const float* __restrict__ W, // gat2_W [80,60]
                                                float* __restrict__ z2) {    // [N,32]
    int tile = blockIdx.x;   // 3125 tiles of 16 rows
    int lane = threadIdx.x;  // wave32, EXEC all ones
    int r = lane & 15;
    int h = lane >> 4;
    int row = tile * 16 + r;
    int j1 = r + 16;                       // cols 16..31
    int j1c = (j1 < EMB) ? j1 : (EMB - 1); // clamped, always in-bounds
    float jmask = (j1 < EMB) ? 1.f : 0.f;  // pad cols 30,31 with 0 via value mask
    v8f c0 = {};
    v8f c1 = {};
    for (int k0 = 0; k0 < HID; k0 += 4) {
        int ka = k0 + 2 * h, kb = ka + 1;
        v2f a;
        a.x = g1[row * HID + ka];
        a.y = g1[row * HID + kb];
        v2f b0, b1;
        b0.x = W[ka * 60 + r];  // cols 0..15 (<30, valid)
        b0.y = W[kb * 60 + r];
        b1.x = W[ka * 60 + j1c] * jmask;  // unconditional load, masked value
        b1.y = W[kb * 60 + j1c] * jmask;
        c0 = __builtin_amdgcn_wmma_f32_16x16x4_f32(false, a, false, b0, (short)0, c0, false, false);
        c1 = __builtin_amdgcn_wmma_f32_16x16x4_f32(false, a, false, b1, (short)0, c1, false, false);
    }
#pragma unroll
    for (int v = 0; v < 8; ++v) {
        int m = v + 8 * h;
        int n = tile * 16 + m;
        z2[n * 32 + r] = c0[v];
        z2[n * 32 + r + 16] = c1[v];
    }
}

// ---------- layer 2: S[j] = sum over edges z2[src[e],j]*alpha[e] ----------
__global__ void k_accum_S(const float* __restrict__ z2, const float* __restrict__ alpha,
                          const int* __restrict__ src, float* __restrict__ S, int E) {
    float acc[EMB];
#pragma unroll
    for (int j = 0; j < EMB; ++j) acc[j] = 0.f;
    for (int e = blockIdx.x * blockDim.x + threadIdx.x; e < E; e += gridDim.x * blockDim.x) {
        float a = alpha[e];
        const float* zr = z2 + (size_t)src[e] * 32;
#pragma unroll
        for (int j = 0; j < EMB; ++j) acc[j] += a * zr[j];
    }
    __shared__ float sh[EMB];
    if (threadIdx.x < EMB) sh[threadIdx.x] = 0.f;
    __syncthreads();
    for (int j = 0; j < EMB; ++j) atomicAdd(&sh[j], acc[j]);
    __syncthreads();
    if (threadIdx.x < EMB) atomicAdd(&S[threadIdx.x], sh[threadIdx.x]);
}

// ---------- final tiny head: image branch + combine + log_softmax ----------
__global__ void k_head(const float* __restrict__ x, const float* __restrict__ vocab,
                       const float* __restrict__ W_lin1, const float* __restrict__ Wc1,
                       const float* __restrict__ Wc2, const float* __restrict__ gat2_b,
                       const float* __restrict__ Wg1, const float* __restrict__ bg1,
                       const float* __restrict__ W2, const float* __restrict__ W3,
                       const float* __restrict__ Sbuf, float* __restrict__ out, int out_size) {
    __shared__ float emb[60];  // [2][30]
    __shared__ float cat[60];  // [a0 | a1]
    __shared__ float hc[70];   // [gr(30) | h2(30) | vocab(10)]
    __shared__ float tv[HID];
    int t = threadIdx.x;
    if (t < 60) {
        int gg = t / EMB, j = t - gg * EMB;
        float acc = 0.f;
        for (int k = 0; k < 4096; ++k) acc += x[gg * 4096 + k] * W_lin1[k * EMB + j];
        emb[t] = acc;
        cat[t] = Sbuf[gg * 32 + j] * (1.0f / N_NODES) + gat2_b[j];  // mean_nodes + b2[0]
    }
    if (t >= 64 && t < 64 + VOCAB_SZ) hc[60 + (t - 64)] = vocab[t - 64];
    __syncthreads();
    if (t < EMB) {  // image conv chain -> hc[30+j]
        float e0 = emb[t], e1 = emb[EMB + t];
        float acc = 0.f;
        for (int c = 0; c < HID; ++c) {
            float hv = Wc1[c * 2 + 0] * e0 + Wc1[c * 2 + 1] * e1;
            hv = (hv >= 0.f) ? hv : CONV_SLOPE * hv;
            acc += Wc2[c] * hv;
        }
        hc[EMB + t] = (acc >= 0.f) ? acc : CONV_SLOPE * acc;
    } else if (t >= 32 && t < 32 + EMB) {  // gr -> hc[j]
        int j = t - 32;
        float acc = bg1[j];
        for (int k = 0; k < 60; ++k) acc += cat[k] * Wg1[k * EMB + j];
        hc[j] = acc;
    }
    __syncthreads();
    if (t < HID) {
        float acc = 0.f;
        for (int k = 0; k < 70; ++k) acc += hc[k] * W2[k * HID + t];
        tv[t] = acc;
    }
    __syncthreads();
    if (t == 0) {
        float l0 = 0.f, l1 = 0.f;
        for (int c = 0; c < HID; ++c) {
            l0 += tv[c] * W3[c * 2 + 0];
            l1 += tv[c] * W3[c * 2 + 1];
        }
        float m = fmaxf(l0, l1);
        float lse = m + logf(expf(l0 - m) + expf(l1 - m));
        out[0] = l0 - lse;
        out[1] = l1 - lse;
        if (out_size > 2) out[2] = 0.f;
    }
}

static inline int nblk(int n, int b) { return (n + b - 1) / b; }

extern "C" void kernel_launch(void* const* d_in, const int* in_sizes, int n_in,
                              void* d_out, int out_size, void* d_ws, size_t ws_size,
                              hipStream_t stream) {
    const float* x        = (const float*)d_in[0];
    const float* vocab    = (const float*)d_in[1];
    const float* node_fts = (const float*)d_in[2];
    const int*   src      = (const int*)d_in[3];
    const int*   dst      = (const int*)d_in[4];
    const float* W_lin1   = (const float*)d_in[5];
    const float* Wc1      = (const float*)d_in[6];
    const float* Wc2      = (const float*)d_in[7];
    const float* gat1_W   = (const float*)d_in[8];
    const float* gat1_al  = (const float*)d_in[9];
    const float* gat1_ar  = (const float*)d_in[10];
    const float* gat1_b   = (const float*)d_in[11];
    const float* gat2_W   = (const float*)d_in[12];
    const float* gat2_al  = (const float*)d_in[13];
    const float* gat2_ar  = (const float*)d_in[14];
    const float* gat2_b   = (const float*)d_in[15];
    const float* Wg1      = (const float*)d_in[16];
    const float* bg1      = (const float*)d_in[17];
    const float* W2       = (const float*)d_in[18];
    const float* W3       = (const float*)d_in[19];

    // workspace layout (floats)
    float*    bufA  = (float*)d_ws;                 // N*80 : z1, later z2 (N*32)
    float*    bufB  = bufA + (size_t)N_NODES * HID; // N*80 : rst1 -> g1
    float*    ebuf  = bufB + (size_t)N_NODES * HID; // E    : e -> ex -> alpha
    float*    elb   = ebuf + (size_t)N_EDGES;       // N
    float*    erb   = elb + N_NODES;                // N
    unsigned* emaxu = (unsigned*)(erb + N_NODES);   // N
    float*    denom = (float*)(emaxu + N_NODES);    // N
    float*    Sbuf  = denom + N_NODES;              // 2*32

    const int B = 256;
    const int totNH = N_NODES * HID;  // 4,000,000
    const int totEH = N_EDGES * HID;  // 64,000,000

    for (int g = 0; g < 2; ++g) {
        const float* nf = node_fts + (size_t)g * N_NODES * 3;
        const int* sg = src + (size_t)g * N_EDGES;
        const int* dg = dst + (size_t)g * N_EDGES;
        float* Sg = Sbuf + g * 32;

        // ----- GAT layer 1 (head 0 only) -----
        k_fill_f32<<<nblk(totNH, B), B, 0, stream>>>(bufB, 0.f, totNH);
        k_fill_u32<<<nblk(N_NODES, B), B, 0, stream>>>(emaxu, 0u, N_NODES);
        k_fill_f32<<<nblk(N_NODES, B), B, 0, stream>>>(denom, 0.f, N_NODES);
        k_z1_wmma<<<N_NODES / 16, 32, 0, stream>>>(nf, gat1_W, bufA);
        k_elr<<<nblk(N_NODES, B), B, 0, stream>>>(bufA, gat1_al, gat1_ar, elb, erb, HID, HID);
        k_edge_e<<<nblk(N_EDGES, B), B, 0, stream>>>(sg, dg, elb, erb, ebuf, emaxu, N_EDGES);
        k_edge_exp<<<nblk(N_EDGES, B), B, 0, stream>>>(dg, ebuf, emaxu, denom, N_EDGES);
        k_alpha<<<nblk(N_EDGES, B), B, 0, stream>>>(dg, ebuf, denom, N_EDGES);
        k_scatter<<<nblk(totEH, B), B, 0, stream>>>(sg, dg, ebuf, bufA, bufB, totEH);
        k_bias_relu<<<nblk(totNH, B), B, 0, stream>>>(bufB, gat1_b, totNH);

        // ----- GAT layer 2 (head 0 only; node-mean folded into edge sum) -----
        k_fill_u32<<<nblk(N_NODES, B), B, 0, stream>>>(emaxu, 0u, N_NODES);
        k_fill_f32<<<nblk(N_NODES, B), B, 0, stream>>>(denom, 0.f, N_NODES);
        k_fill_f32<<<1, 64, 0, stream>>>(Sg, 0.f, 32);
        k_z2_wmma<<<N_NODES / 16, 32, 0, stream>>>(bufB, gat2_W, bufA);
        k_elr<<<nblk(N_NODES, B), B, 0, stream>>>(bufA, gat2_al, gat2_ar, elb, erb, 32, EMB);
        k_edge_e<<<nblk(N_EDGES, B), B, 0, stream>>>(sg, dg, elb, erb, ebuf, emaxu, N_EDGES);
        k_edge_exp<<<nblk(N_EDGES, B), B, 0, stream>>>(dg, ebuf, emaxu, denom, N_EDGES);
        k_alpha<<<nblk(N_EDGES, B), B, 0, stream>>>(dg, ebuf, denom, N_EDGES);
        k_accum_S<<<512, 256, 0, stream>>>(bufA, ebuf, sg, Sg, N_EDGES);
    }

    k_head<<<1, 128, 0, stream>>>(x, vocab, W_lin1, Wc1, Wc2, gat2_b, Wg1, bg1, W2, W3,
                                  Sbuf, (float*)d_out, out_size);
}